// cond_10024453669342
// MI455X (gfx1250) — compile-verified
//
#include <hip/hip_runtime.h>

typedef float v2f __attribute__((ext_vector_type(2)));
typedef float v8f __attribute__((ext_vector_type(8)));

#define K_CODES   256
#define C_DIM     64
#define DHW       110592      /* 48*48*48 */
#define CDHW      7077888     /* 64*DHW */
#define N_ROWS    221184      /* 2*DHW */
#define N_TILES   13824       /* N_ROWS/16 */
#define TOTAL_ELEM 14155776.0f

/* workspace layout (floats) */
#define WS_CNORM  0
#define WS_COUNTS 256
#define WS_LOSS   512
#define WS_DW     768
#define WS_TOTAL  (768 + K_CODES * C_DIM)

/* output layout (floats, tuple flattened in return order) */
#define OUT_Q     0
#define OUT_LOSS  14155776
#define OUT_IDX   14155777
#define OUT_SUM   14376961
#define OUT_EMB   14377217

__global__ void zero_ws_kernel(float* ws, int n) {
    int i = blockIdx.x * blockDim.x + threadIdx.x;
    if (i < n) ws[i] = 0.0f;
}

__global__ void code_norms_kernel(const float* __restrict__ W, float* __restrict__ cnorm) {
    int k = threadIdx.x;                 // one block of 256 threads
    float s = 0.0f;
    #pragma unroll 8
    for (int c = 0; c < C_DIM; ++c) {
        float w = W[k * C_DIM + c];
        s += w * w;
    }
    cnorm[k] = s;
}

__global__ __launch_bounds__(128) void vq_main_kernel(const float* __restrict__ X,
                                                      const float* __restrict__ W,
                                                      float* __restrict__ out,
                                                      float* __restrict__ ws) {
    // Codebook staged in LDS, pre-swizzled into WMMA B-fragment order:
    // element [(tile*16 + t)*32 + lane] (a float2) == W[tile*16+(lane&15)][4t+2*(lane>>4) .. +1]
    __shared__ float ldsW[K_CODES * C_DIM];   // 64 KB exactly

    const float* cnormg  = ws + WS_CNORM;
    float*       counts  = ws + WS_COUNTS;
    float*       lossAcc = ws + WS_LOSS;
    float*       dw      = ws + WS_DW;

    const int tid = threadIdx.x;
    for (int e = tid; e < K_CODES * C_DIM; e += 128) {
        int s    = e >> 6;          // frag-step = tile*16 + t
        int q    = e & 63;          // lane*2 + j
        int ln   = q >> 1, j = q & 1;
        int tile = s >> 4, t = s & 15;
        int code = tile * 16 + (ln & 15);
        int c    = 4 * t + 2 * (ln >> 4) + j;
        ldsW[e] = W[code * C_DIM + c];
    }
    __syncthreads();

    const int lane = tid & 31;
    const int wave = tid >> 5;
    const int m    = lane & 15;
    const int half = lane >> 4;

    // per-lane code norms for codes tile*16 + m
    float cnr[16];
    #pragma unroll
    for (int tile = 0; tile < 16; ++tile) cnr[tile] = cnormg[tile * 16 + m];

    const v2f* fragW = (const v2f*)ldsW;

    for (int j = 0; j < 8; ++j) {
        const int T    = blockIdx.x * 32 + j * 4 + wave;   // 432*32 == 13824 tiles
        const int row0 = T * 16;
        const int b    = row0 / DHW;
        const int sp   = row0 - b * DHW;
        const float* xb = X + (size_t)b * CDHW + sp;
        float*       qb = out + OUT_Q + (size_t)b * CDHW + sp;

        // prefetch next row-tile's 16x64 fp32 block (64 cachelines of 64B)
        if (j < 7) {
            int Tn  = blockIdx.x * 32 + (j + 1) * 4 + wave;
            int rn0 = Tn * 16;
            int bn  = rn0 / DHW;
            int spn = rn0 - bn * DHW;
            const float* xn = X + (size_t)bn * CDHW + spn;
            __builtin_prefetch(xn + (size_t)(2 * lane) * DHW, 0, 1);
            __builtin_prefetch(xn + (size_t)(2 * lane + 1) * DHW, 0, 1);
        }

        // A fragments (16x4 f32 layout) + partial ||x||^2
        v2f a[16];
        float pn = 0.0f;
        #pragma unroll
        for (int t = 0; t < 16; ++t) {
            int c0 = 4 * t + 2 * half;
            float x0 = xb[(size_t)c0 * DHW + m];
            float x1 = xb[(size_t)c0 * DHW + DHW + m];
            a[t].x = x0; a[t].y = x1;
            pn += x0 * x0 + x1 * x1;
        }
        float xnorm = pn + __shfl_xor(pn, 16, 32);   // full row norm, row = lane&15

        float minv[8];
        int   mini[8];
        #pragma unroll
        for (int r = 0; r < 8; ++r) { minv[r] = 3.4e38f; mini[r] = 0; }

        #pragma unroll 2    // two independent accumulator chains for matrix-pipe ILP
        for (int tile = 0; tile < 16; ++tile) {
            v8f acc = {};
            #pragma unroll
            for (int t = 0; t < 16; ++t) {
                v2f bf = fragW[(tile * 16 + t) * 32 + lane];
                acc = __builtin_amdgcn_wmma_f32_16x16x4_f32(
                        false, a[t], false, bf, (short)0, acc, false, false);
            }
            float cn   = cnr[tile];
            int   code = tile * 16 + m;
            #pragma unroll
            for (int r = 0; r < 8; ++r) {
                float d = cn - 2.0f * acc[r];     // ||e||^2 - 2 x.e  (argmin-equivalent)
                if (d < minv[r]) { minv[r] = d; mini[r] = code; }
            }
        }

        // argmin across the 16 lanes of each half (tie-break: smaller index, like jnp.argmin)
        #pragma unroll
        for (int off = 1; off <= 8; off <<= 1) {
            #pragma unroll
            for (int r = 0; r < 8; ++r) {
                float ov = __shfl_xor(minv[r], off, 32);
                int   oi = __shfl_xor(mini[r], off, 32);
                if (ov < minv[r] || (ov == minv[r] && oi < mini[r])) { minv[r] = ov; mini[r] = oi; }
            }
        }

        // distribute per-row winner: lanes 0..15 end with row==lane result
        int sel = lane & 7;
        float myMin = minv[0];
        int   myIdx = mini[0];
        #pragma unroll
        for (int r = 1; r < 8; ++r) { if (sel == r) { myMin = minv[r]; myIdx = mini[r]; } }
        int src = (lane & 7) | ((lane & 8) << 1);   // lanes 8..15 fetch rows 8..15 from upper half
        myMin = __shfl(myMin, src, 32);
        myIdx = __shfl(myIdx, src, 32);

        if (lane < 16) {
            out[OUT_IDX + row0 + lane] = (float)myIdx;
            atomicAdd(&counts[myIdx], 1.0f);
        }
        float lp = (lane < 16) ? (xnorm + myMin) : 0.0f;   // == ||x - e||^2
        #pragma unroll
        for (int off = 16; off >= 1; off >>= 1) lp += __shfl_xor(lp, off, 32);
        if (lane == 0) atomicAdd(lossAcc, lp);

        // quantized write (straight-through: x + (q - x)) + dw segment-sum.
        // Uses the A-fragments still in registers -- no global re-read of X.
        // Lane owns (row m, channels 4t+2*half, 4t+2*half+1); codebook value pair
        // for (code=rowIdx, c0) lives contiguously in frag-ordered LDS.
        int rowIdx = __shfl(myIdx, m, 32);   // broadcast row m's code to both halves
        int base2  = (rowIdx & ~15) * 32 + (rowIdx & 15) + 16 * half;  // v2f index, + 32*t per step
        #pragma unroll
        for (int t = 0; t < 16; ++t) {
            int c0 = 4 * t + 2 * half;
            v2f q = fragW[base2 + 32 * t];
            qb[(size_t)c0 * DHW + m]       = a[t].x + (q.x - a[t].x);
            qb[(size_t)c0 * DHW + DHW + m] = a[t].y + (q.y - a[t].y);
            atomicAdd(&dw[rowIdx * C_DIM + c0],     a[t].x);
            atomicAdd(&dw[rowIdx * C_DIM + c0 + 1], a[t].y);
        }
    }
}

__global__ void finalize_kernel(const float* __restrict__ emaCS,
                                const float* __restrict__ emaW,
                                float* __restrict__ out,
                                const float* __restrict__ ws) {
    __shared__ float red[256];
    const int k = threadIdx.x;
    const float* counts = ws + WS_COUNTS;
    const float* dw     = ws + WS_DW;

    float cs = 0.99f * emaCS[k] + 0.01f * counts[k];
    red[k] = cs;
    __syncthreads();
    for (int s = 128; s > 0; s >>= 1) {
        if (k < s) red[k] += red[k + s];
        __syncthreads();
    }
    float n  = red[0];
    float wk = (cs + 1e-5f) / (n + 256.0f * 1e-5f) * n;

    #pragma unroll 4
    for (int c = 0; c < C_DIM; ++c) {
        float nw = 0.99f * emaW[k * C_DIM + c] + 0.01f * dw[k * C_DIM + c];
        out[OUT_EMB + k * C_DIM + c] = nw / wk;
    }
    out[OUT_SUM + k] = counts[k];
    if (k == 0) out[OUT_LOSS] = 2.5f * ws[WS_LOSS] / TOTAL_ELEM;  // 0.25 * mean * 10
}

extern "C" void kernel_launch(void* const* d_in, const int* in_sizes, int n_in,
                              void* d_out, int out_size, void* d_ws, size_t ws_size,
                              hipStream_t stream) {
    const float* X     = (const float*)d_in[0];   // [2,64,48,48,48]
    const float* W     = (const float*)d_in[1];   // [256,64]
    const float* emaCS = (const float*)d_in[2];   // [256]
    const float* emaW  = (const float*)d_in[3];   // [256,64]
    float* out = (float*)d_out;
    float* ws  = (float*)d_ws;

    zero_ws_kernel<<<(WS_TOTAL + 255) / 256, 256, 0, stream>>>(ws, WS_TOTAL);
    code_norms_kernel<<<1, 256, 0, stream>>>(W, ws + WS_CNORM);
    vq_main_kernel<<<N_TILES / 32, 128, 0, stream>>>(X, W, out, ws);
    finalize_kernel<<<1, 256, 0, stream>>>(emaCS, emaW, out, ws);
}